// NGCF_34127810134331
// MI455X (gfx1250) — compile-verified
//
#include <hip/hip_runtime.h>
#include <hip/hip_bf16.h>

#define N_USER   100000
#define N_ITEM   200000
#define N_NODES  (N_USER + N_ITEM)
#define EMB      64
#define BATCH    4096
#define NSMALL   (3 * BATCH)     // 12288 rows
#define OUTC     256             // 4 * EMB concatenated
#define SLOT_NONE 0x7FFFFFFF

typedef float v2f __attribute__((ext_vector_type(2)));
typedef float v8f __attribute__((ext_vector_type(8)));

// ---------------------------------------------------------------- init
__global__ __launch_bounds__(256) void ngcf_init_kernel(int* __restrict__ node2slot,
                                                        float* __restrict__ side_small) {
    int i = blockIdx.x * 256 + threadIdx.x;
    if (i < N_NODES) node2slot[i] = SLOT_NONE;
    if (i < NSMALL * EMB) side_small[i] = 0.0f;
}

// ---------------------------------------------------------------- mark needed nodes
__global__ __launch_bounds__(256) void ngcf_mark_kernel(const int* __restrict__ users,
                                                        const int* __restrict__ pos,
                                                        const int* __restrict__ neg,
                                                        int* __restrict__ node2slot) {
    int s = blockIdx.x * 256 + threadIdx.x;
    if (s >= NSMALL) return;
    int node;
    if (s < BATCH)          node = users[s];
    else if (s < 2 * BATCH) node = N_USER + pos[s - BATCH];
    else                    node = N_USER + neg[s - 2 * BATCH];
    atomicMin(&node2slot[node], s);
}

// ---------------------------------------------------------------- filtered edge scatter
// One edge per wave32. 'rep' is wave-uniform -> no divergence inside a wave.
// Only ~4% of the 4M edges pass the filter, so traffic is ~48MB index scan
// + ~512B per passing edge instead of the naive ~2GB.
__global__ __launch_bounds__(256) void ngcf_scatter_kernel(const int* __restrict__ rowi,
                                                           const int* __restrict__ coli,
                                                           const float* __restrict__ vals,
                                                           const int* __restrict__ node2slot,
                                                           const float* __restrict__ user_emb,
                                                           const float* __restrict__ item_emb,
                                                           float* __restrict__ side_small,
                                                           int nnz) {
    int e    = blockIdx.x * 8 + (threadIdx.x >> 5);
    int lane = threadIdx.x & 31;
    if (e >= nnz) return;
    int rep = node2slot[rowi[e]];
    if (rep == SLOT_NONE) return;                 // dead edge: destination row unused
    int c   = coli[e];
    float v = vals[e];
    const float* eb = (c < N_USER) ? (user_emb + (size_t)c * EMB)
                                   : (item_emb + (size_t)(c - N_USER) * EMB);
    float* ob = side_small + (size_t)rep * EMB;
    unsafeAtomicAdd(&ob[lane],      v * eb[lane]);
    unsafeAtomicAdd(&ob[lane + 32], v * eb[lane + 32]);
}

// ---------------------------------------------------------------- gather small_side / ego / out[:, 0:64]
__global__ __launch_bounds__(256) void ngcf_gather_kernel(const int* __restrict__ users,
                                                          const int* __restrict__ pos,
                                                          const int* __restrict__ neg,
                                                          const int* __restrict__ node2slot,
                                                          const float* __restrict__ side_small,
                                                          const float* __restrict__ user_emb,
                                                          const float* __restrict__ item_emb,
                                                          float* __restrict__ small_side,
                                                          float* __restrict__ ego_buf,
                                                          float* __restrict__ out) {
    int i = blockIdx.x * 256 + threadIdx.x;
    if (i >= NSMALL * EMB) return;
    int s = i >> 6;
    int d = i & 63;
    int node;
    float e;
    if (s < BATCH) {
        node = users[s];
        e = user_emb[(size_t)node * EMB + d];
    } else if (s < 2 * BATCH) {
        int it = pos[s - BATCH];
        node = N_USER + it;
        e = item_emb[(size_t)it * EMB + d];
    } else {
        int it = neg[s - 2 * BATCH];
        node = N_USER + it;
        e = item_emb[(size_t)it * EMB + d];
    }
    int rep = node2slot[node];                    // representative slot holds the full sum
    small_side[i] = side_small[(size_t)rep * EMB + d];
    ego_buf[i]    = e;
    out[(size_t)s * OUTC + d] = e;                // layer-0 ego, unnormalized
}

// ---------------------------------------------------------------- fused WMMA layer
// One wave owns a 16-row strip. Dual accumulator chains:
//   accS = small_side @ W_gc + b_gc          (A = side fragments)
//   accB = (ego .* small_side) @ W_bi + b_bi (A = elementwise product fragments)
// then leaky_relu, wave-shuffle L2 row norm, writes new ego + normalized output cols.
__global__ __launch_bounds__(256) void ngcf_layer_kernel(const float* __restrict__ small_side,
                                                         float* __restrict__ ego_buf,
                                                         const float* __restrict__ Wg,
                                                         const float* __restrict__ bg,
                                                         const float* __restrict__ Wb,
                                                         const float* __restrict__ bb,
                                                         float* __restrict__ out,
                                                         int col_base) {
    __shared__ float sWg[EMB * EMB];
    __shared__ float sWb[EMB * EMB];
    __shared__ float sbg[EMB];
    __shared__ float sbb[EMB];

    const int tid = threadIdx.x;
    for (int i = tid; i < EMB * EMB; i += 256) { sWg[i] = Wg[i]; sWb[i] = Wb[i]; }
    if (tid < EMB) { sbg[tid] = bg[tid]; sbb[tid] = bb[tid]; }
    __syncthreads();

    const int wave = tid >> 5;
    const int lane = tid & 31;
    const int m    = lane & 15;      // A-matrix row within strip (lanes 0-15 / 16-31 both map M=0..15)
    const int kh   = lane >> 4;      // which K-pair this lane holds
    const int nidx = lane & 15;      // B/C column within 16-wide tile
    const int row0 = (blockIdx.x * 8 + wave) * 16;

    // A fragments: 16x4 f32 layout -> lane holds K = 4*kt + 2*kh + {0,1}
    v2f a_frag[16], e_frag[16];
    const float* srow = small_side + (size_t)(row0 + m) * EMB + 2 * kh;
    const float* erow = ego_buf    + (size_t)(row0 + m) * EMB + 2 * kh;
#pragma unroll
    for (int kt = 0; kt < 16; ++kt) {
        v2f a = *(const v2f*)(srow + 4 * kt);
        v2f e = *(const v2f*)(erow + 4 * kt);
        a_frag[kt] = a;
        e_frag[kt] = v2f{a.x * e.x, a.y * e.y};
    }

    float eg[4][8];
#pragma unroll
    for (int nt = 0; nt < 4; ++nt) {
        const int ncol = nt * 16 + nidx;
        float bgv = sbg[ncol];
        float bbv = sbb[ncol];
        v8f accS = {bgv, bgv, bgv, bgv, bgv, bgv, bgv, bgv};   // bias pre-loaded as C
        v8f accB = {bbv, bbv, bbv, bbv, bbv, bbv, bbv, bbv};
#pragma unroll
        for (int kt = 0; kt < 16; ++kt) {
            const int krow = 4 * kt + 2 * kh;
            v2f bgf, bbf;
            bgf.x = sWg[(krow + 0) * EMB + ncol];
            bgf.y = sWg[(krow + 1) * EMB + ncol];
            bbf.x = sWb[(krow + 0) * EMB + ncol];
            bbf.y = sWb[(krow + 1) * EMB + ncol];
            accS = __builtin_amdgcn_wmma_f32_16x16x4_f32(false, a_frag[kt], false, bgf,
                                                         (short)0, accS, false, false);
            accB = __builtin_amdgcn_wmma_f32_16x16x4_f32(false, e_frag[kt], false, bbf,
                                                         (short)0, accB, false, false);
        }
#pragma unroll
        for (int r = 0; r < 8; ++r) {
            float v = accS[r] + accB[r];
            eg[nt][r] = (v >= 0.0f) ? v : 0.2f * v;   // leaky_relu slope 0.2
        }
    }

    // per-row sum of squares; row r+8*kh lives entirely in this lane's 16-lane half,
    // so xor-shuffles with masks 1,2,4,8 complete the reduction.
    float ss[8];
#pragma unroll
    for (int r = 0; r < 8; ++r) {
        float s = 0.0f;
#pragma unroll
        for (int nt = 0; nt < 4; ++nt) s += eg[nt][r] * eg[nt][r];
        ss[r] = s;
    }
#pragma unroll
    for (int mask = 1; mask <= 8; mask <<= 1) {
#pragma unroll
        for (int r = 0; r < 8; ++r) ss[r] += __shfl_xor(ss[r], mask, 32);
    }

#pragma unroll
    for (int r = 0; r < 8; ++r) {
        const int row = row0 + r + 8 * kh;            // C/D layout: VGPR r -> M = r + 8*half
        const float scale = 1.0f / fmaxf(sqrtf(ss[r]), 1e-12f);
#pragma unroll
        for (int nt = 0; nt < 4; ++nt) {
            const int n = nt * 16 + nidx;
            ego_buf[(size_t)row * EMB + n] = eg[nt][r];                    // next layer's ego
            out[(size_t)row * OUTC + col_base + n] = eg[nt][r] * scale;    // normalized concat slice
        }
    }
}

// ---------------------------------------------------------------- launch
extern "C" void kernel_launch(void* const* d_in, const int* in_sizes, int n_in,
                              void* d_out, int out_size, void* d_ws, size_t ws_size,
                              hipStream_t stream) {
    const int*   users    = (const int*)d_in[0];
    const int*   pos      = (const int*)d_in[1];
    const int*   neg      = (const int*)d_in[2];
    const int*   rowi     = (const int*)d_in[3];
    const int*   coli     = (const int*)d_in[4];
    const float* vals     = (const float*)d_in[5];
    const float* user_emb = (const float*)d_in[6];
    const float* item_emb = (const float*)d_in[7];
    const float* W_gc     = (const float*)d_in[8];
    const float* b_gc     = (const float*)d_in[9];
    const float* W_bi     = (const float*)d_in[10];
    const float* b_bi     = (const float*)d_in[11];
    float* out = (float*)d_out;
    const int nnz = in_sizes[3];

    // workspace carve-up (~11 MB)
    char* ws = (char*)d_ws;
    int*   node2slot  = (int*)ws;
    size_t off = ((size_t)N_NODES * sizeof(int) + 255) & ~(size_t)255;
    float* side_small = (float*)(ws + off);                 // 12288 x 64 compact accumulator
    float* small_side = side_small + (size_t)NSMALL * EMB;  // gathered per-slot side
    float* ego_buf    = small_side + (size_t)NSMALL * EMB;  // evolving ego

    const int initN = (NSMALL * EMB > N_NODES) ? NSMALL * EMB : N_NODES;
    ngcf_init_kernel<<<(initN + 255) / 256, 256, 0, stream>>>(node2slot, side_small);
    ngcf_mark_kernel<<<(NSMALL + 255) / 256, 256, 0, stream>>>(users, pos, neg, node2slot);
    ngcf_scatter_kernel<<<(nnz + 7) / 8, 256, 0, stream>>>(rowi, coli, vals, node2slot,
                                                           user_emb, item_emb, side_small, nnz);
    ngcf_gather_kernel<<<(NSMALL * EMB) / 256, 256, 0, stream>>>(users, pos, neg, node2slot,
                                                                 side_small, user_emb, item_emb,
                                                                 small_side, ego_buf, out);
    for (int k = 0; k < 3; ++k) {
        ngcf_layer_kernel<<<NSMALL / 128, 256, 0, stream>>>(small_side, ego_buf,
                                                            W_gc + (size_t)k * EMB * EMB,
                                                            b_gc + (size_t)k * EMB,
                                                            W_bi + (size_t)k * EMB * EMB,
                                                            b_bi + (size_t)k * EMB,
                                                            out, EMB * (k + 1));
    }
}